// AdapterRouter_635655160027
// MI455X (gfx1250) — compile-verified
//
#include <hip/hip_runtime.h>
#include <hip/hip_bf16.h>
#include <math.h>

// Cosine-similarity argmax: q(256) vs E(N x 256), N = 1,000,000.
// Memory-bound (~1 GB streamed, ~44us floor at 23.3 TB/s).
//  - Dot products via V_WMMA_F32_16X16X4_F32 (full f32, bit-faithful math);
//    query replicated across B's 16 columns -> each wave does 16 row-dots.
//  - Global->LDS staging via GLOBAL_LOAD_ASYNC_TO_LDS_B128 (ASYNCcnt),
//    double-buffered so each wave always has ~16KB in flight.

typedef __attribute__((ext_vector_type(2))) float v2f;
typedef __attribute__((ext_vector_type(8))) float v8f;

#define DIM        256
#define ROW_STRIDE 260   // floats; 1040B rows: 16B-aligned, stride%64==4 -> no bank conflicts
#define TILE_ROWS  16
#define TILE_BYTES (TILE_ROWS * ROW_STRIDE * 4)   // 16640
#define EPSV       1e-8f

__device__ __forceinline__ unsigned int floatKey(float f) {
    unsigned int b = __float_as_uint(f);
    return b ^ ((b & 0x80000000u) ? 0xFFFFFFFFu : 0x80000000u);
}

__global__ void cosine_init_kernel(unsigned long long* ws) {
    if (threadIdx.x == 0 && blockIdx.x == 0) ws[0] = 0ull;
}

// Issue 32 async b128 copies: one contiguous 16KB global tile -> padded LDS tile.
// gp0 = tile base + lane*16B; lbase = LDS byte offset of buffer + lane*16.
__device__ __forceinline__ void stage_tile_async(const float* gp0, unsigned lbase) {
    #pragma unroll
    for (int it = 0; it < 32; ++it) {
        // global byte offset: it*512 (tile is contiguous); LDS: padded rows
        unsigned laddr = lbase + (unsigned)((it >> 1) * 1040 + (it & 1) * 512);
        asm volatile("global_load_async_to_lds_b128 %0, %1, off offset:%c2"
                     :: "v"(laddr), "v"(gp0), "i"(it * 512)
                     : "memory");
    }
}

__global__ __launch_bounds__(32)
void cosine_topk_kernel(const float* __restrict__ q,
                        const float* __restrict__ emb,
                        int n_rows,
                        unsigned long long* __restrict__ ws) {
    __shared__ __align__(16) float lds_q[DIM];                           // normalized query
    __shared__ __align__(16) float lds_tile[2 * TILE_ROWS * ROW_STRIDE]; // double buffer

    const int lane = threadIdx.x;          // 0..31 (wave32)
    const int half = lane >> 4;            // 0: lanes 0-15, 1: lanes 16-31
    const int r    = lane & 15;

    // ---- Normalize the query once per block: qn = q / (||q|| + eps) ----
    float4 qa = *(const float4*)(q + lane * 8);
    float4 qb = *(const float4*)(q + lane * 8 + 4);
    float qs = qa.x*qa.x + qa.y*qa.y + qa.z*qa.z + qa.w*qa.w
             + qb.x*qb.x + qb.y*qb.y + qb.z*qb.z + qb.w*qb.w;
    #pragma unroll
    for (int off = 16; off >= 1; off >>= 1) qs += __shfl_xor(qs, off, 32);
    const float qscale = 1.0f / (sqrtf(qs) + EPSV);
    qa.x *= qscale; qa.y *= qscale; qa.z *= qscale; qa.w *= qscale;
    qb.x *= qscale; qb.y *= qscale; qb.z *= qscale; qb.w *= qscale;
    *(float4*)(lds_q + lane * 8)     = qa;
    *(float4*)(lds_q + lane * 8 + 4) = qb;
    __syncthreads();

    const int numTiles = (n_rows + TILE_ROWS - 1) / TILE_ROWS;

    float bestScore = -INFINITY;
    int   bestIdx   = 0;

    const float2*  qn2      = (const float2*)lds_q;
    const unsigned ldsBase0 = (unsigned)(size_t)(void*)lds_tile + (unsigned)(lane * 16);

    // ---- Prologue: kick off the first tile's async copy into buffer 0 ----
    int tile = blockIdx.x;
    if (tile < numTiles)
        stage_tile_async(emb + (size_t)tile * TILE_ROWS * DIM + lane * 4, ldsBase0);

    int buf = 0;
    for (; tile < numTiles; tile += gridDim.x) {
        const int rowBase = tile * TILE_ROWS;
        const int nxt     = tile + (int)gridDim.x;

        // ---- Stage next tile into the other buffer while we compute ----
        if (nxt < numTiles) {
            asm volatile("s_wait_dscnt 0x0" ::: "memory");   // prior reads of that buffer done
            stage_tile_async(emb + (size_t)nxt * TILE_ROWS * DIM + lane * 4,
                             ldsBase0 + (unsigned)((buf ^ 1) * TILE_BYTES));
            // L2 prefetch two tiles ahead
            const int pf = nxt + (int)gridDim.x;
            if (pf < numTiles) {
                const char* np = (const char*)(emb + (size_t)pf * TILE_ROWS * DIM);
                #pragma unroll
                for (int p = 0; p < 4; ++p)
                    __builtin_prefetch(np + (size_t)(p * 32 + lane) * 128, 0, 1);
            }
            asm volatile("s_wait_asynccnt 0x20" ::: "memory"); // current buffer complete
        } else {
            asm volatile("s_wait_asynccnt 0x0" ::: "memory");
        }

        const float* tb = lds_tile + buf * (TILE_ROWS * ROW_STRIDE);

        // ---- Per-row sum of squares: each lane-half reads half a row ----
        const float2* rp = (const float2*)&tb[r * ROW_STRIDE + half * 128];
        float ss = 0.0f;
        #pragma unroll
        for (int k = 0; k < 64; ++k) {
            float2 f = rp[k];
            ss += f.x * f.x + f.y * f.y;
        }
        ss += __shfl_xor(ss, 16, 32);   // lane L now holds rowss(L & 15)

        // ---- 64 x V_WMMA_F32_16X16X4_F32 accumulate the 256-d dots ----
        v8f acc = {};
        const float2* tile2 = (const float2*)tb;
        #pragma unroll 8
        for (int c = 0; c < 64; ++c) {
            // A 16x4: lane half h supplies K = {2h, 2h+1} of row r
            v2f a, b;
            float2 af = tile2[r * (ROW_STRIDE / 2) + c * 2 + half];
            a[0] = af.x; a[1] = af.y;
            // B 4x16: query chunk replicated across all 16 columns
            float2 bf = qn2[c * 2 + half];
            b[0] = bf.x; b[1] = bf.y;
            acc = __builtin_amdgcn_wmma_f32_16x16x4_f32(
                    /*neg_a=*/false, a, /*neg_b=*/false, b,
                    /*c_mod=*/(short)0, acc,
                    /*reuse_a=*/false, /*reuse_b=*/false);
        }

        // ---- Scale by 1/(||e||+eps), track local best ----
        #pragma unroll
        for (int v = 0; v < 8; ++v) {
            const int rowInTile = v + 8 * half;
            const int grow = rowBase + rowInTile;
            float rss = __shfl(ss, rowInTile, 32);
            float sim = acc[v] / (sqrtf(rss) + EPSV);
            if (grow < n_rows && sim > bestScore) {
                bestScore = sim;
                bestIdx   = grow;
            }
        }

        buf ^= 1;
    }

    // ---- Wave-wide (score,index) max, then one global atomic ----
    unsigned long long packed =
        ((unsigned long long)floatKey(bestScore) << 32) | (unsigned int)bestIdx;
    #pragma unroll
    for (int off = 16; off >= 1; off >>= 1) {
        unsigned long long other = __shfl_xor(packed, off, 32);
        if (other > packed) packed = other;
    }
    if (lane == 0) atomicMax(ws, packed);
}

__global__ void cosine_finalize_kernel(const unsigned long long* __restrict__ ws,
                                       float* __restrict__ out) {
    if (threadIdx.x == 0 && blockIdx.x == 0) {
        unsigned long long p = ws[0];
        unsigned int key = (unsigned int)(p >> 32);
        unsigned int idx = (unsigned int)(p & 0xFFFFFFFFu);
        unsigned int bits = (key & 0x80000000u) ? (key ^ 0x80000000u) : ~key;
        out[0] = (float)idx;                 // best_match
        out[1] = __uint_as_float(bits);      // best_score
    }
}

extern "C" void kernel_launch(void* const* d_in, const int* in_sizes, int n_in,
                              void* d_out, int out_size, void* d_ws, size_t ws_size,
                              hipStream_t stream) {
    const float* q   = (const float*)d_in[0];
    const float* emb = (const float*)d_in[1];
    const int n_rows = in_sizes[1] / DIM;    // 1,000,000

    unsigned long long* ws = (unsigned long long*)d_ws;
    float* out = (float*)d_out;

    cosine_init_kernel<<<1, 1, 0, stream>>>(ws);

    const int numTiles = (n_rows + TILE_ROWS - 1) / TILE_ROWS;
    int blocks = numTiles < 4096 ? numTiles : 4096;
    cosine_topk_kernel<<<blocks, 32, 0, stream>>>(q, emb, n_rows, ws);

    cosine_finalize_kernel<<<1, 1, 0, stream>>>(ws, out);
}